// SELECT_fusion_block_1580547973805
// MI455X (gfx1250) — compile-verified
//
#include <hip/hip_runtime.h>
#include <hip/hip_bf16.h>

typedef __attribute__((ext_vector_type(2))) float v2f;
typedef __attribute__((ext_vector_type(4))) float v4f;
typedef __attribute__((ext_vector_type(8))) float v8f;

#define NT     64
#define CTOT   256
#define FOLDC  32
#define HH     56
#define WW     56
#define HWSZ   (HH * WW)          // 3136 = 196 * 16
#define KDIM   96
#define TSEG   8

// ---------------------------------------------------------------------------
// Kernel 1: 3x3 stride-1 maxpool (pad=1) over the first 32 channels -> sel(ws)
// ---------------------------------------------------------------------------
__global__ __launch_bounds__(256) void pool_kernel(const float* __restrict__ x,
                                                   float* __restrict__ sel) {
  int idx = blockIdx.x * 256 + threadIdx.x;
  const int total = NT * FOLDC * HWSZ;
  if (idx >= total) return;
  int pix = idx % HWSZ;
  int pc  = idx / HWSZ;            // nt*FOLDC + c
  int c   = pc % FOLDC;
  int nt  = pc / FOLDC;
  int h = pix / WW, w = pix % WW;
  const float* p = x + ((size_t)nt * CTOT + c) * HWSZ;
  int h0 = (h > 0) ? h - 1 : h, h1 = (h < HH - 1) ? h + 1 : h;
  int w0 = (w > 0) ? w - 1 : w, w1 = (w < WW - 1) ? w + 1 : w;
  float m = p[pix];
  for (int hh = h0; hh <= h1; ++hh)
    for (int ww2 = w0; ww2 <= w1; ++ww2)
      m = fmaxf(m, p[hh * WW + ww2]);
  sel[idx] = m;
}

// ---------------------------------------------------------------------------
// Kernel 2: fused 96->32 channel mix via V_WMMA_F32_16X16X4_F32.
// One wave = one 16-pixel tile of one frame. M=16 pixels, N=32 (2 halves),
// K=96 (24 chunks of 4). Bias folded into the C accumulator.
// ---------------------------------------------------------------------------
__global__ __launch_bounds__(128) void fused_kernel(const float* __restrict__ x,
                                                    const float* __restrict__ Wm,
                                                    const float* __restrict__ bias,
                                                    const float* __restrict__ sel,
                                                    float* __restrict__ out) {
  __shared__ float sW[FOLDC * KDIM];      // W row-major [o][c], 12 KB
  __shared__ float sA[4][KDIM * 16];      // per-wave cat tile [c][m], 24 KB

  const int lane = threadIdx.x & 31;
  const int wave = threadIdx.x >> 5;
  const int nt   = blockIdx.y;
  const int t    = nt & (TSEG - 1);
  const int tile = blockIdx.x * 4 + wave; // 0..195
  const int pixb = tile * 16;

  // Cooperative load of W into LDS (3072 floats / 128 threads)
  for (int i = threadIdx.x; i < FOLDC * KDIM; i += 128) sW[i] = Wm[i];

  // Stage cat = [left(sel t-1) | part | right(sel t+1)] for this tile.
  const float* partp = x + (size_t)nt * CTOT * HWSZ;            // ch 0..31 = part
  const float* selm  = (t > 0)        ? sel + (size_t)(nt - 1) * FOLDC * HWSZ : nullptr;
  const float* selp  = (t < TSEG - 1) ? sel + (size_t)(nt + 1) * FOLDC * HWSZ : nullptr;

  float* sAw = &sA[wave][0];
  const int m = lane & 15;          // pixel within tile / N column
  const int chalf = lane >> 4;      // 0 or 1
  for (int cc = 0; cc < KDIM; cc += 2) {
    int c = cc + chalf;
    float v;
    if (c < FOLDC)            v = selm ? selm[c * HWSZ + pixb + m] : 0.0f;
    else if (c < 2 * FOLDC)   v = partp[(c - FOLDC) * HWSZ + pixb + m];
    else                      v = selp ? selp[(c - 2 * FOLDC) * HWSZ + pixb + m] : 0.0f;
    sAw[c * 16 + m] = v;
  }
  __syncthreads();

  // Accumulators seeded with bias (all 8 rows of a lane share column N).
  float b0 = bias[m];
  float b1 = bias[16 + m];
  v8f acc0 = {b0, b0, b0, b0, b0, b0, b0, b0};
  v8f acc1 = {b1, b1, b1, b1, b1, b1, b1, b1};

  // A layout (16x4 f32): lanes 0-15 hold K = k0+{0,1}, lanes 16-31 K = k0+{2,3}.
  // B layout (4x16 f32): same K split; N = lane&15 (+16 for second half).
  const int kshift = chalf * 2;
  for (int k0 = 0; k0 < KDIM; k0 += 4) {
    int kk = k0 + kshift;
    v2f a;  a.x  = sAw[kk * 16 + m];        a.y  = sAw[(kk + 1) * 16 + m];
    v2f bm0; bm0.x = sW[m * KDIM + kk];       bm0.y = sW[m * KDIM + kk + 1];
    v2f bm1; bm1.x = sW[(m + 16) * KDIM + kk]; bm1.y = sW[(m + 16) * KDIM + kk + 1];
    acc0 = __builtin_amdgcn_wmma_f32_16x16x4_f32(false, a, false, bm0,
                                                 (short)0, acc0, false, false);
    acc1 = __builtin_amdgcn_wmma_f32_16x16x4_f32(false, a, false, bm1,
                                                 (short)0, acc1, false, false);
  }

  // D layout: VGPR r -> pixel = pixb + r + 8*chalf, column N = m (+16).
  // Per lane the 8 results are 8 consecutive pixels -> two b128 stores each.
  float* outp = out + (size_t)nt * CTOT * HWSZ;
  const int mb = chalf * 8;
  v4f lo0 = {acc0[0], acc0[1], acc0[2], acc0[3]};
  v4f hi0 = {acc0[4], acc0[5], acc0[6], acc0[7]};
  v4f lo1 = {acc1[0], acc1[1], acc1[2], acc1[3]};
  v4f hi1 = {acc1[4], acc1[5], acc1[6], acc1[7]};
  *(v4f*)(outp + (size_t)m * HWSZ + pixb + mb)            = lo0;
  *(v4f*)(outp + (size_t)m * HWSZ + pixb + mb + 4)        = hi0;
  *(v4f*)(outp + (size_t)(m + 16) * HWSZ + pixb + mb)     = lo1;
  *(v4f*)(outp + (size_t)(m + 16) * HWSZ + pixb + mb + 4) = hi1;
}

// ---------------------------------------------------------------------------
// Kernel 3: pass-through copy of channels 32..255 (b128 vectorized)
// ---------------------------------------------------------------------------
__global__ __launch_bounds__(256) void copy_kernel(const float* __restrict__ x,
                                                   float* __restrict__ out) {
  const int per_frame4 = (CTOT - FOLDC) * HWSZ / 4;   // 175616 float4 per frame
  long long i = (long long)blockIdx.x * 256 + threadIdx.x;
  if (i >= (long long)NT * per_frame4) return;
  int frame = (int)(i / per_frame4);
  int rem   = (int)(i % per_frame4);
  size_t off = (size_t)frame * CTOT * HWSZ + (size_t)FOLDC * HWSZ + (size_t)rem * 4;
  *(v4f*)(out + off) = *(const v4f*)(x + off);
}

extern "C" void kernel_launch(void* const* d_in, const int* in_sizes, int n_in,
                              void* d_out, int out_size, void* d_ws, size_t ws_size,
                              hipStream_t stream) {
  const float* x  = (const float*)d_in[0];   // (64, 256, 56, 56)
  const float* Wm = (const float*)d_in[1];   // (32, 96)
  const float* b  = (const float*)d_in[2];   // (32,)
  float* out = (float*)d_out;
  float* sel = (float*)d_ws;                 // needs 64*32*3136*4 = ~25.7 MB

  // 1) maxpool -> sel
  {
    int total = NT * FOLDC * HWSZ;
    pool_kernel<<<(total + 255) / 256, 256, 0, stream>>>(x, sel);
  }
  // 2) WMMA fused channel mix -> out[:, :32]
  {
    dim3 grid(HWSZ / 16 / 4, NT);            // (49, 64); 4 waves per block
    fused_kernel<<<grid, 128, 0, stream>>>(x, Wm, b, sel, out);
  }
  // 3) passthrough -> out[:, 32:]
  {
    long long total4 = (long long)NT * (CTOT - FOLDC) * HWSZ / 4;
    copy_kernel<<<(int)((total4 + 255) / 256), 256, 0, stream>>>(x, out);
  }
}